// AttnReadout_16655883174583
// MI455X (gfx1250) — compile-verified
//
#include <hip/hip_runtime.h>
#include <hip/hip_bf16.h>
#include <math.h>
#include <stdint.h>

typedef float v2f __attribute__((ext_vector_type(2)));
typedef float v8f __attribute__((ext_vector_type(8)));

#define D128 128

// ---------------- zero scratch ----------------
__global__ void zero_kernel(float* p, int n) {
    int i = blockIdx.x * blockDim.x + threadIdx.x;
    if (i < n) p[i] = 0.f;
}

// ---------------- column stats (sum, sumsq) ----------------
__global__ __launch_bounds__(128) void stats_kernel(const float* __restrict__ feat,
                                                    float* __restrict__ sums,
                                                    float* __restrict__ sumsq,
                                                    int rows_per_block) {
    int d = threadIdx.x;
    size_t r0 = (size_t)blockIdx.x * rows_per_block;
    float s = 0.f, s2 = 0.f;
    for (int r = 0; r < rows_per_block; ++r) {
        float x = feat[(r0 + r) * D128 + d];
        s += x;
        s2 = fmaf(x, x, s2);
    }
    atomicAdd(&sums[d], s);
    atomicAdd(&sumsq[d], s2);
}

// ---------------- finalize BN coefficients: y = x*colA + colB ----------------
__global__ void finalize_stats(const float* __restrict__ sums, const float* __restrict__ sumsq,
                               const float* __restrict__ gamma, const float* __restrict__ beta,
                               float* __restrict__ colA, float* __restrict__ colB, float invN) {
    int d = threadIdx.x;
    float mean = sums[d] * invN;
    float var  = sumsq[d] * invN - mean * mean;
    float inv  = rsqrtf(var + 1e-5f);
    float a    = inv * gamma[d];
    colA[d] = a;
    colB[d] = beta[d] - mean * a;
}

// ---------------- per-segment additive vector: combo[b][d] ----------------
__global__ __launch_bounds__(128) void combo_kernel(const float* __restrict__ intend,
                                                    const float* __restrict__ feat,
                                                    const int* __restrict__ last_nodes,
                                                    const float* __restrict__ Wv,
                                                    const float* __restrict__ bv,
                                                    const float* __restrict__ Wi,
                                                    const float* __restrict__ bi,
                                                    const float* __restrict__ colA,
                                                    const float* __restrict__ colB,
                                                    float* __restrict__ combo) {
    int b = blockIdx.x, d = threadIdx.x;
    __shared__ float ln[D128];
    __shared__ float it[D128];
    int lastr = last_nodes[b];
    ln[d] = fmaf(feat[(size_t)lastr * D128 + d], colA[d], colB[d]);
    it[d] = intend[(size_t)b * D128 + d];
    __syncthreads();
    float acc = bv[d] + bi[d];
    const float* wvr = Wv + (size_t)d * D128;
    const float* wir = Wi + (size_t)d * D128;
    for (int k = 0; k < D128; ++k)
        acc = fmaf(it[k], wvr[k], fmaf(ln[k], wir[k], acc));
    combo[(size_t)b * D128 + d] = acc;
}

// ---------------- main WMMA GEMM (feat_u = featn @ Wu.T) fused with gate e1 ----------------
// 128 threads = 4 waves (wave32); each wave owns a 16-row stripe -> 64 rows/block.
// Wu (64 KB) staged via async global->LDS copies (ASYNCcnt); feat tile (32 KB)
// normalized and staged with coalesced float4 loads; WMMA fed entirely from LDS.
__global__ __launch_bounds__(128) void gemm_gate_kernel(const float* __restrict__ feat,
                                                        const float* __restrict__ Wu,
                                                        const float* __restrict__ We,
                                                        const int* __restrict__ segids,
                                                        const float* __restrict__ combo,
                                                        const float* __restrict__ colA,
                                                        const float* __restrict__ colB,
                                                        float* __restrict__ e1out) {
    __shared__ float sWu[D128 * D128];  // 64 KB: B-matrix B[k][n] = Wu[n][k]
    __shared__ float sF[64 * D128];     // 32 KB: normalized 64x128 feat tile
    __shared__ float sWe[D128];
    __shared__ float e1part[64];

    int tid = threadIdx.x;

    // --- issue async copy of Wu into LDS (tracked by ASYNCcnt) ---
    {
        const float4* Wu4 = (const float4*)Wu;
        float4* sWu4 = (float4*)sWu;
        #pragma unroll
        for (int it = 0; it < 32; ++it) {
            int idx = it * 128 + tid;  // 4096 float4s total
            uint64_t gaddr = (uint64_t)(uintptr_t)(const void*)(Wu4 + idx);
            uint32_t laddr = (uint32_t)(uintptr_t)(void*)(sWu4 + idx);
            asm volatile("global_load_async_to_lds_b128 %0, %1, off"
                         :: "v"(laddr), "v"(gaddr) : "memory");
        }
    }

    // --- meanwhile: normalize + stage 64x128 feat tile, coalesced ---
    {
        const float4* colA4 = (const float4*)colA;
        const float4* colB4 = (const float4*)colB;
        int cg = tid & 31;          // fixed column-group (4 cols) per thread
        float4 a4 = colA4[cg];
        float4 b4 = colB4[cg];
        const float4* feat4 = (const float4*)(feat + (size_t)blockIdx.x * 64 * D128);
        float4* sF4 = (float4*)sF;
        #pragma unroll
        for (int it = 0; it < 16; ++it) {
            int row = it * 4 + (tid >> 5);
            int idx = row * 32 + cg;
            float4 x = feat4[idx];
            float4 y;
            y.x = fmaf(x.x, a4.x, b4.x);
            y.y = fmaf(x.y, a4.y, b4.y);
            y.z = fmaf(x.z, a4.z, b4.z);
            y.w = fmaf(x.w, a4.w, b4.w);
            sF4[idx] = y;
        }
    }
    sWe[tid] = We[tid];
    if (tid < 64) e1part[tid] = 0.f;

    // wait for this wave's async copies, then block barrier
    asm volatile("s_wait_asynccnt 0" ::: "memory");
    __syncthreads();

    int wave = tid >> 5;
    int lane = tid & 31;
    int m0   = blockIdx.x * 64 + wave * 16;
    int mrow = wave * 16 + (lane & 15);  // local row in sF for A fragment
    int khalf = (lane >> 4) << 1;        // lanes 16-31 hold K=k0+2,k0+3
    int nlow = lane & 15;                // N for B/C fragments

    v8f acc[8] = {};                     // 8 x (16x16 f32) tiles = 16x128 stripe

    const float* frow = sF + mrow * D128;

    for (int k0 = 0; k0 < D128; k0 += 4) {
        int kb = k0 + khalf;
        v2f a;
        a.x = frow[kb];
        a.y = frow[kb + 1];
        #pragma unroll
        for (int t = 0; t < 8; ++t) {
            int n = t * 16 + nlow;
            v2f bfr;
            bfr.x = sWu[n * D128 + kb];       // B[kb][n]   = Wu[n][kb]
            bfr.y = sWu[n * D128 + kb + 1];   // B[kb+1][n] = Wu[n][kb+1]
            acc[t] = __builtin_amdgcn_wmma_f32_16x16x4_f32(
                false, a, false, bfr, (short)0, acc[t], false, false);
        }
    }

    // Epilogue: sigmoid(feat_u + combo[seg]) dot We, reduced per row via LDS atomics.
    // C layout: VGPR j, lanes 0-15 -> M=j, lanes 16-31 -> M=j+8; N = lane%16 (+16t).
    int mhi = (lane >> 4) << 3;  // 0 or 8
    int sgr[8];
    #pragma unroll
    for (int j = 0; j < 8; ++j) sgr[j] = segids[m0 + mhi + j];
    #pragma unroll
    for (int t = 0; t < 8; ++t) {
        int n = t * 16 + nlow;
        float wn = sWe[n];
        #pragma unroll
        for (int j = 0; j < 8; ++j) {
            float pre = acc[t][j] + combo[(size_t)sgr[j] * D128 + n];
            float sg  = 1.f / (1.f + __expf(-pre));
            atomicAdd(&e1part[wave * 16 + mhi + j], sg * wn);
        }
    }
    __syncthreads();
    if (tid < 64) e1out[(size_t)blockIdx.x * 64 + tid] = e1part[tid];
}

// ---------------- segment softmax + weighted segment sums -> outputs ----------------
__global__ __launch_bounds__(128) void segment_out_kernel(const float* __restrict__ feat,
                                                          const float* __restrict__ pw,
                                                          const int* __restrict__ segids,
                                                          const float* __restrict__ e1,
                                                          const float* __restrict__ colA,
                                                          const float* __restrict__ colB,
                                                          float* __restrict__ out,
                                                          int Ntot, int Btot) {
    int b = blockIdx.x;
    int tid = threadIdx.x;

    // segment_ids are sorted: binary search for [s, e)
    int lo = 0, hi = Ntot;
    while (lo < hi) { int mid = (lo + hi) >> 1; if (segids[mid] < b) lo = mid + 1; else hi = mid; }
    int s = lo;
    hi = Ntot;
    while (lo < hi) { int mid = (lo + hi) >> 1; if (segids[mid] < b + 1) lo = mid + 1; else hi = mid; }
    int e = lo;

    __shared__ float red[128];
    __shared__ float sm[2];

    // pass 1: segment max
    float m = -INFINITY;
    for (int r = s + tid; r < e; r += 128) m = fmaxf(m, e1[r]);
    red[tid] = m;
    __syncthreads();
    for (int off = 64; off > 0; off >>= 1) {
        if (tid < off) red[tid] = fmaxf(red[tid], red[tid + off]);
        __syncthreads();
    }
    if (tid == 0) sm[0] = red[0];
    __syncthreads();
    m = sm[0];
    __syncthreads();

    // pass 2: sum of exp
    float ssum = 0.f;
    for (int r = s + tid; r < e; r += 128) ssum += __expf(e1[r] - m);
    red[tid] = ssum;
    __syncthreads();
    for (int off = 64; off > 0; off >>= 1) {
        if (tid < off) red[tid] += red[tid + off];
        __syncthreads();
    }
    if (tid == 0) sm[1] = (red[0] > 0.f) ? 1.f / red[0] : 0.f;
    __syncthreads();
    float inv_denom = sm[1];

    // pass 3: thread owns column d; coalesced row sweeps
    int d = tid;
    float a  = colA[d];
    float bb = colB[d];
    float acc = 0.f, accp = 0.f;
    for (int r = s; r < e; ++r) {
        if (r + 4 < e) __builtin_prefetch(&feat[(size_t)(r + 4) * D128 + d], 0, 0);
        float alpha = __expf(e1[r] - m) * inv_denom;
        float x = fmaf(feat[(size_t)r * D128 + d], a, bb);
        acc  = fmaf(x, alpha, acc);
        accp = fmaf(x, pw[r], accp);
    }
    out[(size_t)b * D128 + d] = acc;
    out[(size_t)Btot * D128 + (size_t)b * D128 + d] = accp;
}

extern "C" void kernel_launch(void* const* d_in, const int* in_sizes, int n_in,
                              void* d_out, int out_size, void* d_ws, size_t ws_size,
                              hipStream_t stream) {
    const float* feat    = (const float*)d_in[0];
    const float* intend  = (const float*)d_in[1];
    const float* pw      = (const float*)d_in[2];
    const int*   lastn   = (const int*)d_in[3];
    const int*   segids  = (const int*)d_in[4];
    const float* gamma   = (const float*)d_in[5];
    const float* beta    = (const float*)d_in[6];
    const float* Wu      = (const float*)d_in[7];
    const float* Wv      = (const float*)d_in[8];
    const float* bv      = (const float*)d_in[9];
    const float* Wi      = (const float*)d_in[10];
    const float* bi      = (const float*)d_in[11];
    const float* We      = (const float*)d_in[12];
    float* out = (float*)d_out;

    int N = in_sizes[0] / D128;   // 524288
    int B = in_sizes[1] / D128;   // 4096

    float* ws    = (float*)d_ws;
    float* sums  = ws;                 // 128
    float* sumsq = ws + 128;           // 128
    float* colA  = ws + 256;           // 128
    float* colB  = ws + 384;           // 128
    float* combo = ws + 512;           // B*128
    float* e1    = combo + (size_t)B * D128;  // N

    zero_kernel<<<1, 256, 0, stream>>>(sums, 256);

    int stat_blocks = 1024;
    stats_kernel<<<stat_blocks, 128, 0, stream>>>(feat, sums, sumsq, N / stat_blocks);

    finalize_stats<<<1, 128, 0, stream>>>(sums, sumsq, gamma, beta, colA, colB, 1.0f / (float)N);

    combo_kernel<<<B, 128, 0, stream>>>(intend, feat, lastn, Wv, bv, Wi, bi, colA, colB, combo);

    gemm_gate_kernel<<<N / 64, 128, 0, stream>>>(feat, Wu, We, segids, combo, colA, colB, e1);

    segment_out_kernel<<<B, 128, 0, stream>>>(feat, pw, segids, e1, colA, colB, out, N, B);
}